// GINJumpingKnowledge_79869211837073
// MI455X (gfx1250) — compile-verified
//
#include <hip/hip_runtime.h>
#include <hip/hip_bf16.h>

// ---------------------------------------------------------------------------
// GIN + JumpingKnowledge for MI455X (gfx1250, wave32, WMMA)
//
// Pipeline per layer l (L=3):
//   agg = h + scatter_add(h[src] -> dst)            (edge kernel, f32 atomics, L2-resident)
//   m   = relu(BN(agg @ W1[l] + b1[l]))             (WMMA f32 16x16x4 GEMM, fused epilogue)
//   h   = relu(m @ W2[l] + b2[l])                   (WMMA f32 16x16x4 GEMM, fused epilogue)
//   pooled[:, l*128:(l+1)*128] += segment_sum(h)    (atomic pool)
// Tail:
//   z      = (pooled / counts) @ Wp + bp            (WMMA, K=384)
//   logits = z @ Wc + bc                            (tiny VALU kernel)
// ---------------------------------------------------------------------------

#define NN      100000
#define NE      1600000
#define NG      256
#define DD      128
#define NL      3
#define PROJ    128
#define NCLS    10
#define JKD     (NL * DD)      // 384
#define BN_EPS  1e-5f

typedef __attribute__((ext_vector_type(2))) float v2f;
typedef __attribute__((ext_vector_type(8))) float v8f;

// D = A(16x4) * B(4x16) + C(16x16), all fp32. 8-arg form:
// (neg_a, A, neg_b, B, c_mod, C, reuse_a, reuse_b)
__device__ __forceinline__ v8f wmma4(v2f a, v2f b, v8f c) {
    return __builtin_amdgcn_wmma_f32_16x16x4_f32(
        false, a, false, b, (short)0, c, false, false);
}

// --------------------------- utility kernels -------------------------------

__global__ __launch_bounds__(256) void zero_kernel(float* __restrict__ p, int n) {
    int i = blockIdx.x * 256 + threadIdx.x;
    if (i < n) p[i] = 0.0f;
}

__global__ __launch_bounds__(256) void count_kernel(const int* __restrict__ batch,
                                                    float* __restrict__ counts) {
    int i = blockIdx.x * 256 + threadIdx.x;
    if (i < NN) atomicAdd(&counts[batch[i]], 1.0f);
}

// agg = h   (vectorized copy; edge kernel accumulates on top)
__global__ __launch_bounds__(256) void copy_kernel(const float* __restrict__ src,
                                                   float* __restrict__ dst) {
    long long i = (long long)blockIdx.x * 256 + threadIdx.x;   // float4 index
    if (i < (long long)NN * DD / 4)
        ((float4*)dst)[i] = ((const float4*)src)[i];
}

// one wave per edge; lane moves float4 (16B) -> 4 f32 atomics into agg[dst]
__global__ __launch_bounds__(256) void edge_kernel(const int* __restrict__ ei,
                                                   const float* __restrict__ h,
                                                   float* __restrict__ agg) {
    long long t = (long long)blockIdx.x * 256 + threadIdx.x;
    int e    = (int)(t >> 5);
    int lane = (int)(t & 31);
    if (e >= NE) return;
    int s = ei[e];          // edge_index[0, e] (src)
    int d = ei[NE + e];     // edge_index[1, e] (dst)
    float4 v = *(const float4*)&h[(long long)s * DD + lane * 4];
    float* p = &agg[(long long)d * DD + lane * 4];
    atomicAdd(p + 0, v.x);
    atomicAdd(p + 1, v.y);
    atomicAdd(p + 2, v.z);
    atomicAdd(p + 3, v.w);
}

// ----------------------------- WMMA GEMMs ----------------------------------
// Each wave computes a 16-row x 128-col strip of the output.
// A fragment (16x4 f32): lanes 0-15 carry K={0,1}, lanes 16-31 carry K={2,3},
// row = lane&15.  B fragment (4x16): v[0]=W[k][col], v[1]=W[k+1][col] with
// k = kbase + 2*(lane>>4), col = ct*16 + (lane&15).
// C/D layout: acc[ct][r] is row = row_base + r + 8*(lane>>4), col = ct*16+(lane&15).

// GEMM1: m = relu( BN( A @ W1 + b1 ) )
__global__ __launch_bounds__(256) void gemm1_bn_relu_kernel(
    const float* __restrict__ A,      // [NN,128]
    const float* __restrict__ W,      // [128,128]
    const float* __restrict__ b1,
    const float* __restrict__ bng, const float* __restrict__ bnb,
    const float* __restrict__ bnm, const float* __restrict__ bnv,
    float* __restrict__ M)            // [NN,128]
{
    int wave = threadIdx.x >> 5;
    int lane = threadIdx.x & 31;
    int strip = blockIdx.x * 8 + wave;
    int row_base = strip * 16;
    if (row_base >= NN) return;
    int hl = lane >> 4;               // 0/1 -> K pair select
    int ml = lane & 15;
    const float* arow = A + (long long)(row_base + ml) * DD;

    v8f acc[8];
    v8f zero = {0.f, 0.f, 0.f, 0.f, 0.f, 0.f, 0.f, 0.f};
#pragma unroll
    for (int t = 0; t < 8; ++t) acc[t] = zero;

    for (int kb = 0; kb < DD / 4; ++kb) {
        int k = kb * 4 + hl * 2;
        float2 av = *(const float2*)&arow[k];
        v2f a; a.x = av.x; a.y = av.y;
#pragma unroll
        for (int ct = 0; ct < 8; ++ct) {
            v2f b;
            b.x = W[k * DD + ct * 16 + ml];
            b.y = W[(k + 1) * DD + ct * 16 + ml];
            acc[ct] = wmma4(a, b, acc[ct]);
        }
    }

    int rrow0 = row_base + hl * 8;
#pragma unroll
    for (int ct = 0; ct < 8; ++ct) {
        int col = ct * 16 + ml;
        float sc = bng[col] * rsqrtf(bnv[col] + BN_EPS);
        float sh = bnb[col] + (b1[col] - bnm[col]) * sc;
#pragma unroll
        for (int r = 0; r < 8; ++r) {
            float v = acc[ct][r] * sc + sh;
            M[(long long)(rrow0 + r) * DD + col] = fmaxf(v, 0.0f);
        }
    }
}

// GEMM2: h = relu( M @ W2 + b2 )
__global__ __launch_bounds__(256) void gemm2_relu_kernel(
    const float* __restrict__ M, const float* __restrict__ W,
    const float* __restrict__ b2, float* __restrict__ H)
{
    int wave = threadIdx.x >> 5;
    int lane = threadIdx.x & 31;
    int strip = blockIdx.x * 8 + wave;
    int row_base = strip * 16;
    if (row_base >= NN) return;
    int hl = lane >> 4;
    int ml = lane & 15;
    const float* arow = M + (long long)(row_base + ml) * DD;

    v8f acc[8];
    v8f zero = {0.f, 0.f, 0.f, 0.f, 0.f, 0.f, 0.f, 0.f};
#pragma unroll
    for (int t = 0; t < 8; ++t) acc[t] = zero;

    for (int kb = 0; kb < DD / 4; ++kb) {
        int k = kb * 4 + hl * 2;
        float2 av = *(const float2*)&arow[k];
        v2f a; a.x = av.x; a.y = av.y;
#pragma unroll
        for (int ct = 0; ct < 8; ++ct) {
            v2f b;
            b.x = W[k * DD + ct * 16 + ml];
            b.y = W[(k + 1) * DD + ct * 16 + ml];
            acc[ct] = wmma4(a, b, acc[ct]);
        }
    }

    int rrow0 = row_base + hl * 8;
#pragma unroll
    for (int ct = 0; ct < 8; ++ct) {
        int col = ct * 16 + ml;
        float bias = b2[col];
#pragma unroll
        for (int r = 0; r < 8; ++r) {
            float v = acc[ct][r] + bias;
            H[(long long)(rrow0 + r) * DD + col] = fmaxf(v, 0.0f);
        }
    }
}

// segment-sum pooling: pooled[g, layer*128 + c] += h[n, c]
__global__ __launch_bounds__(256) void pool_kernel(const float* __restrict__ H,
                                                   const int* __restrict__ batch,
                                                   float* __restrict__ pooled,
                                                   int layer) {
    long long t = (long long)blockIdx.x * 256 + threadIdx.x;
    int n    = (int)(t >> 5);
    int lane = (int)(t & 31);
    if (n >= NN) return;
    int g = batch[n];
    float4 v = *(const float4*)&H[(long long)n * DD + lane * 4];
    float* p = &pooled[(long long)g * JKD + layer * DD + lane * 4];
    atomicAdd(p + 0, v.x);
    atomicAdd(p + 1, v.y);
    atomicAdd(p + 2, v.z);
    atomicAdd(p + 3, v.w);
}

// z = (pooled / counts) @ Wp + bp  -- rows=256, K=384, N=128.
// one wave per block; block b handles rows [16b, 16b+16)
__global__ __launch_bounds__(32) void proj_kernel(const float* __restrict__ pooled,
                                                  const float* __restrict__ counts,
                                                  const float* __restrict__ Wp,
                                                  const float* __restrict__ bp,
                                                  float* __restrict__ Z) {
    int lane = threadIdx.x & 31;
    int hl = lane >> 4;
    int ml = lane & 15;
    int row_base = blockIdx.x * 16;
    int arow = row_base + ml;
    float inv = 1.0f / fmaxf(counts[arow], 1.0f);
    const float* ap = pooled + (long long)arow * JKD;

    v8f acc[8];
    v8f zero = {0.f, 0.f, 0.f, 0.f, 0.f, 0.f, 0.f, 0.f};
#pragma unroll
    for (int t = 0; t < 8; ++t) acc[t] = zero;

    for (int kb = 0; kb < JKD / 4; ++kb) {
        int k = kb * 4 + hl * 2;
        float2 av = *(const float2*)&ap[k];
        v2f a; a.x = av.x * inv; a.y = av.y * inv;  // fold mean-pool divide
#pragma unroll
        for (int ct = 0; ct < 8; ++ct) {
            v2f b;
            b.x = Wp[k * PROJ + ct * 16 + ml];
            b.y = Wp[(k + 1) * PROJ + ct * 16 + ml];
            acc[ct] = wmma4(a, b, acc[ct]);
        }
    }

    int rrow0 = row_base + hl * 8;
#pragma unroll
    for (int ct = 0; ct < 8; ++ct) {
        int col = ct * 16 + ml;
        float bias = bp[col];
#pragma unroll
        for (int r = 0; r < 8; ++r)
            Z[(long long)(rrow0 + r) * PROJ + col] = acc[ct][r] + bias;
    }
}

// logits = z @ Wc + bc   (256 x 10, trivial)
__global__ __launch_bounds__(256) void logits_kernel(const float* __restrict__ Z,
                                                     const float* __restrict__ Wc,
                                                     const float* __restrict__ bc,
                                                     float* __restrict__ out) {
    int g = blockIdx.x * 256 + threadIdx.x;
    if (g >= NG) return;
    const float* zr = Z + (long long)g * PROJ;
#pragma unroll
    for (int c = 0; c < NCLS; ++c) {
        float acc = bc[c];
        for (int k = 0; k < PROJ; ++k) acc += zr[k] * Wc[k * NCLS + c];
        out[g * NCLS + c] = acc;
    }
}

// ------------------------------- launcher ----------------------------------

extern "C" void kernel_launch(void* const* d_in, const int* in_sizes, int n_in,
                              void* d_out, int out_size, void* d_ws, size_t ws_size,
                              hipStream_t stream) {
    const float* x    = (const float*)d_in[0];
    const int*   ei   = (const int*)d_in[1];
    const int*   batch= (const int*)d_in[2];
    const float* W1   = (const float*)d_in[3];   // [L,128,128]
    const float* b1   = (const float*)d_in[4];   // [L,128]
    const float* bng  = (const float*)d_in[5];
    const float* bnb  = (const float*)d_in[6];
    const float* bnm  = (const float*)d_in[7];
    const float* bnv  = (const float*)d_in[8];
    const float* W2   = (const float*)d_in[9];
    const float* b2   = (const float*)d_in[10];
    const float* Wp   = (const float*)d_in[11];  // [384,128]
    const float* bp   = (const float*)d_in[12];
    const float* Wc   = (const float*)d_in[13];  // [128,10]
    const float* bc   = (const float*)d_in[14];

    // workspace layout
    const long long nodef = (long long)NN * DD;                 // 12.8M floats
    size_t need = (size_t)(2 * nodef + NG * JKD + NG) * sizeof(float);
    if (ws_size < need) return;
    float* buf0   = (float*)d_ws;
    float* buf1   = buf0 + nodef;
    float* pooled = buf1 + nodef;            // [256, 384]
    float* counts = pooled + NG * JKD;       // [256]

    float* zout = (float*)d_out;             // [256,128]
    float* lout = zout + NG * PROJ;          // [256,10]

    // init: zero pooled + counts, then per-graph node counts
    {
        int n = NG * JKD + NG;
        zero_kernel<<<(n + 255) / 256, 256, 0, stream>>>(pooled, n);
        count_kernel<<<(NN + 255) / 256, 256, 0, stream>>>(batch, counts);
    }

    const int gemm_blocks = (NN / 16 + 7) / 8;                      // 782
    const int edge_blocks = (int)(((long long)NE * 32 + 255) / 256);// 200000
    const int copy_blocks = (int)((nodef / 4 + 255) / 256);
    const int pool_blocks = (int)(((long long)NN * 32 + 255) / 256);

    const float* hin = x;
    float* t0 = buf0;   // agg / h destination
    float* t1 = buf1;   // mlp intermediate
    for (int l = 0; l < NL; ++l) {
        // agg = hin + scatter_add
        copy_kernel<<<copy_blocks, 256, 0, stream>>>(hin, t0);
        edge_kernel<<<edge_blocks, 256, 0, stream>>>(ei, hin, t0);
        // m = relu(BN(agg @ W1 + b1))
        gemm1_bn_relu_kernel<<<gemm_blocks, 256, 0, stream>>>(
            t0, W1 + (long long)l * DD * DD, b1 + l * DD,
            bng + l * DD, bnb + l * DD, bnm + l * DD, bnv + l * DD, t1);
        // h = relu(m @ W2 + b2)
        gemm2_relu_kernel<<<gemm_blocks, 256, 0, stream>>>(
            t1, W2 + (long long)l * DD * DD, b2 + l * DD, t0);
        // pooled[:, l*128:] += segment_sum(h)
        pool_kernel<<<pool_blocks, 256, 0, stream>>>(t0, batch, pooled, l);

        hin = t0;
        float* tmp = t0; t0 = t1; t1 = tmp;
    }

    // z = (pooled / counts) @ Wp + bp
    proj_kernel<<<NG / 16, 32, 0, stream>>>(pooled, counts, Wp, bp, zout);
    // logits = z @ Wc + bc
    logits_kernel<<<1, 256, 0, stream>>>(zout, Wc, bc, lout);
}